// RGCN_40200893891113
// MI455X (gfx1250) — compile-verified
//
#include <hip/hip_runtime.h>
#include <hip/hip_bf16.h>

#define NN 100000
#define DD 128
#define EE 1600000
#define LN_EPS 1e-5f

typedef __attribute__((ext_vector_type(2))) float v2f;
typedef __attribute__((ext_vector_type(8))) float v8f;

// ---------------- utility kernels ----------------

__global__ void __launch_bounds__(256) rgcn_zero_f4(float* __restrict__ p, long long n4) {
    long long i = (long long)blockIdx.x * blockDim.x + threadIdx.x;
    if (i < n4) ((float4*)p)[i] = make_float4(0.f, 0.f, 0.f, 0.f);
}

__global__ void __launch_bounds__(256) rgcn_count_int(const int* __restrict__ idx,
                                                      int* __restrict__ cnt, int n) {
    int i = blockIdx.x * blockDim.x + threadIdx.x;
    if (i < n) atomicAdd(cnt + idx[i], 1);
}

__global__ void __launch_bounds__(256) rgcn_cnt_to_rsqrt(const int* __restrict__ cnt,
                                                         float* __restrict__ rs, int n) {
    int i = blockIdx.x * blockDim.x + threadIdx.x;
    if (i < n) rs[i] = rsqrtf((float)max(cnt[i], 1));
}

// Single-workgroup chunked exclusive scan: rowptr[i] = sum(cnt[0..i-1]), rowptr[n] = total.
// Also writes cursor[i] = rowptr[i] for the CSR fill pass.
__global__ void __launch_bounds__(1024) rgcn_scan(const int* __restrict__ cnt,
                                                  int* __restrict__ rowptr,
                                                  int* __restrict__ cursor, int n) {
    __shared__ int buf[1024];
    __shared__ int carry;
    if (threadIdx.x == 0) carry = 0;
    __syncthreads();
    for (int base = 0; base < n; base += 1024) {
        int i = base + (int)threadIdx.x;
        int v = (i < n) ? cnt[i] : 0;
        buf[threadIdx.x] = v;
        __syncthreads();
        for (int off = 1; off < 1024; off <<= 1) {
            int t = (threadIdx.x >= (unsigned)off) ? buf[threadIdx.x - off] : 0;
            __syncthreads();
            buf[threadIdx.x] += t;
            __syncthreads();
        }
        int excl = buf[threadIdx.x] - v + carry;
        if (i < n) { rowptr[i] = excl; cursor[i] = excl; }
        __syncthreads();
        if (threadIdx.x == 1023) carry += buf[1023];
        __syncthreads();
    }
    if (threadIdx.x == 0) rowptr[n] = carry;
}

__global__ void __launch_bounds__(256) rgcn_csr_fill(const int* __restrict__ src,
                                                     const int* __restrict__ dst,
                                                     int* __restrict__ cursor,
                                                     int* __restrict__ csr_src, int n) {
    int e = blockIdx.x * blockDim.x + threadIdx.x;
    if (e < n) {
        int pos = atomicAdd(cursor + dst[e], 1);
        csr_src[pos] = src[e];
    }
}

// Transpose all 8 weight matrices (2 layers x 4 etypes) : Wt[m][n][k] = W[m][k][n]
__global__ void __launch_bounds__(256) rgcn_transpose8(const float* __restrict__ W0,
                                                       const float* __restrict__ W1,
                                                       float* __restrict__ Wt) {
    int idx = blockIdx.x * 256 + threadIdx.x;     // over 8*128*128 = 131072
    int m = idx >> 14;
    int r = (idx >> 7) & 127;                     // output row  (= n of W)
    int c = idx & 127;                            // output col  (= k of W)
    const float* Wsrc = (m < 4) ? (W0 + (long long)m * DD * DD)
                                : (W1 + (long long)(m - 4) * DD * DD);
    Wt[idx] = Wsrc[(long long)c * DD + r];
}

// ---------------- WMMA GEMM: H[N x 128] = rs_out[:,None] * (X @ W) ----------------
// One block = 8 wave32; each wave owns one 16-wide N tile of a shared 16-row M tile.
// Inner loop: one b64 A load + one b64 B load (from transposed W) + one
// V_WMMA_F32_16X16X4_F32. Row scaling folded into the epilogue.

__global__ void __launch_bounds__(256) rgcn_gemm_wmma(const float* __restrict__ X,
                                                      const float* __restrict__ rs,
                                                      const float* __restrict__ Wt,
                                                      float* __restrict__ H) {
    const int lane  = threadIdx.x & 31;
    const int wave  = threadIdx.x >> 5;           // 0..7 -> N tile
    const int m0    = blockIdx.x * 16;
    const int n0    = wave * 16;
    const int mlane = lane & 15;
    const int khalf = (lane >> 4) * 2;            // lanes 0-15: K+0/1, lanes 16-31: K+2/3
    const int row   = m0 + mlane;
    const int col   = n0 + mlane;

    const float* __restrict__ arow = X  + (long long)row * DD + khalf;
    const float* __restrict__ brow = Wt + (long long)col * DD + khalf;

    v8f c = {};
#pragma unroll 8
    for (int k = 0; k < DD; k += 4) {
        v2f a = *(const v2f*)(arow + k);
        v2f b = *(const v2f*)(brow + k);
        c = __builtin_amdgcn_wmma_f32_16x16x4_f32(false, a, false, b, (short)0, c,
                                                  false, false);
    }

    const int rbase = m0 + (lane >> 4) * 8;       // C/D layout: lane half selects M+8
#pragma unroll
    for (int j = 0; j < 8; ++j)
        H[(long long)(rbase + j) * DD + col] = c[j] * rs[rbase + j];
}

// ---------------- fused dst-centric gather + combine + LayerNorm ----------------
// One wave32 per node; 4 channels per lane. Registers accumulate both relations'
// in-edge sums (no atomics), then scale/bias/ReLU/sum + LayerNorm via butterfly
// shuffle reduction.

__global__ void __launch_bounds__(256) rgcn_gather_ln(
        const float* __restrict__ h0, const int* __restrict__ rp0,
        const int* __restrict__ cs0, const float* __restrict__ rsin0,
        const float* __restrict__ bias0,
        const float* __restrict__ h1, const int* __restrict__ rp1,
        const int* __restrict__ cs1, const float* __restrict__ rsin1,
        const float* __restrict__ bias1,
        const float* __restrict__ gamma, const float* __restrict__ beta,
        float* __restrict__ out, int n, int act) {
    int node = blockIdx.x * 8 + (threadIdx.x >> 5);
    if (node >= n) return;
    const int lane = threadIdx.x & 31;

    float a0x = 0.f, a0y = 0.f, a0z = 0.f, a0w = 0.f;
    float a1x = 0.f, a1y = 0.f, a1z = 0.f, a1w = 0.f;

    {
        int jb = rp0[node], je = rp0[node + 1];
        int j = jb;
        for (; j + 1 < je; j += 2) {
            int s0 = cs0[j], s1 = cs0[j + 1];
            float4 v0 = ((const float4*)(h0 + (long long)s0 * DD))[lane];
            float4 v1 = ((const float4*)(h0 + (long long)s1 * DD))[lane];
            a0x += v0.x + v1.x; a0y += v0.y + v1.y;
            a0z += v0.z + v1.z; a0w += v0.w + v1.w;
        }
        if (j < je) {
            float4 v0 = ((const float4*)(h0 + (long long)cs0[j] * DD))[lane];
            a0x += v0.x; a0y += v0.y; a0z += v0.z; a0w += v0.w;
        }
    }
    {
        int jb = rp1[node], je = rp1[node + 1];
        int j = jb;
        for (; j + 1 < je; j += 2) {
            int s0 = cs1[j], s1 = cs1[j + 1];
            float4 v0 = ((const float4*)(h1 + (long long)s0 * DD))[lane];
            float4 v1 = ((const float4*)(h1 + (long long)s1 * DD))[lane];
            a1x += v0.x + v1.x; a1y += v0.y + v1.y;
            a1z += v0.z + v1.z; a1w += v0.w + v1.w;
        }
        if (j < je) {
            float4 v0 = ((const float4*)(h1 + (long long)cs1[j] * DD))[lane];
            a1x += v0.x; a1y += v0.y; a1z += v0.z; a1w += v0.w;
        }
    }

    const float r0 = rsin0[node];
    const float r1 = rsin1[node];
    float4 b0 = ((const float4*)(bias0))[lane];
    float4 b1 = ((const float4*)(bias1))[lane];

    float u0 = a0x * r0 + b0.x, u1 = a0y * r0 + b0.y;
    float u2 = a0z * r0 + b0.z, u3 = a0w * r0 + b0.w;
    float w0 = a1x * r1 + b1.x, w1 = a1y * r1 + b1.y;
    float w2 = a1z * r1 + b1.z, w3 = a1w * r1 + b1.w;
    if (act) {
        u0 = fmaxf(u0, 0.f); u1 = fmaxf(u1, 0.f);
        u2 = fmaxf(u2, 0.f); u3 = fmaxf(u3, 0.f);
        w0 = fmaxf(w0, 0.f); w1 = fmaxf(w1, 0.f);
        w2 = fmaxf(w2, 0.f); w3 = fmaxf(w3, 0.f);
    }
    float s0 = u0 + w0, s1 = u1 + w1, s2 = u2 + w2, s3 = u3 + w3;

    float sum = s0 + s1 + s2 + s3;
    float sq  = s0 * s0 + s1 * s1 + s2 * s2 + s3 * s3;
#pragma unroll
    for (int off = 16; off > 0; off >>= 1) {
        sum += __shfl_xor(sum, off, 32);
        sq  += __shfl_xor(sq,  off, 32);
    }
    const float mean = sum * (1.0f / DD);
    const float var  = sq * (1.0f / DD) - mean * mean;
    const float inv  = rsqrtf(var + LN_EPS);

    float4 g  = ((const float4*)(gamma))[lane];
    float4 bt = ((const float4*)(beta))[lane];
    float4 o;
    o.x = (s0 - mean) * inv * g.x + bt.x;
    o.y = (s1 - mean) * inv * g.y + bt.y;
    o.z = (s2 - mean) * inv * g.z + bt.z;
    o.w = (s3 - mean) * inv * g.w + bt.w;
    ((float4*)(out + (long long)node * DD))[lane] = o;
}

// ---------------- host orchestration ----------------

extern "C" void kernel_launch(void* const* d_in, const int* in_sizes, int n_in,
                              void* d_out, int out_size, void* d_ws, size_t ws_size,
                              hipStream_t stream) {
    const float* x_A = (const float*)d_in[0];
    const float* x_B = (const float*)d_in[1];
    const float* W0  = (const float*)d_in[2];
    const float* b0  = (const float*)d_in[3];
    const float* W1  = (const float*)d_in[4];
    const float* b1  = (const float*)d_in[5];
    const float* lng = (const float*)d_in[6];
    const float* lnb = (const float*)d_in[7];
    // etype order: 0=aa, 1=ab, 2=ba, 3=bb
    const int* src[4] = {(const int*)d_in[8],  (const int*)d_in[10],
                         (const int*)d_in[12], (const int*)d_in[14]};
    const int* dst[4] = {(const int*)d_in[9],  (const int*)d_in[11],
                         (const int*)d_in[13], (const int*)d_in[15]};

    const long long ND = (long long)NN * DD;

    // workspace layout (all 16B-aligned segment boundaries)
    int*   cnt    = (int*)d_ws;                         // [8][N]   (2e=src hist, 2e+1=dst hist)
    float* rs     = (float*)(cnt + 8LL * NN);           // [8][N]   rsqrt(max(deg,1))
    int*   rowptr = (int*)(rs + 8LL * NN);              // [4][N+1]
    int*   cursor = rowptr + 4LL * (NN + 1);            // [4][N]
    int*   csrc   = cursor + 4LL * NN;                  // [4][E]
    float* Wt     = (float*)(csrc + 4LL * EE);          // [8][128][128] transposed weights
    float* h0     = Wt + 8LL * DD * DD;                 // [N][D]
    float* h1     = h0 + ND;
    float* hA1    = h1 + ND;
    float* hB1    = hA1 + ND;

    // ---- degrees + CSR (rebuilt every call; deterministic work) ----
    rgcn_zero_f4<<<(2 * NN + 255) / 256, 256, 0, stream>>>((float*)cnt, 2LL * NN); // 8N ints
    for (int e = 0; e < 4; ++e) {
        rgcn_count_int<<<(EE + 255) / 256, 256, 0, stream>>>(src[e], cnt + (long long)(2*e)   * NN, EE);
        rgcn_count_int<<<(EE + 255) / 256, 256, 0, stream>>>(dst[e], cnt + (long long)(2*e+1) * NN, EE);
    }
    rgcn_cnt_to_rsqrt<<<(8 * NN + 255) / 256, 256, 0, stream>>>(cnt, rs, 8 * NN);
    for (int e = 0; e < 4; ++e) {
        rgcn_scan<<<1, 1024, 0, stream>>>(cnt + (long long)(2*e+1) * NN,
                                          rowptr + (long long)e * (NN + 1),
                                          cursor + (long long)e * NN, NN);
        rgcn_csr_fill<<<(EE + 255) / 256, 256, 0, stream>>>(src[e], dst[e],
                                                            cursor + (long long)e * NN,
                                                            csrc + (long long)e * EE, EE);
    }
    rgcn_transpose8<<<(8 * DD * DD) / 256, 256, 0, stream>>>(W0, W1, Wt);

    float* outA_final = (float*)d_out;
    float* outB_final = (float*)d_out + ND;

    const int gemm_grid = NN / 16;          // 6250
    const int node_grid = (NN + 7) / 8;     // 12500

    // one (layer, dst-ntype) pass: relations e0 (src xs0) and e1 (src xs1)
    auto conv_pair = [&](const float* xs0, int e0, const float* xs1, int e1,
                         int layer, const float* bl,
                         const float* gam, const float* bet,
                         float* out, int act) {
        rgcn_gemm_wmma<<<gemm_grid, 256, 0, stream>>>(
            xs0, rs + (long long)(2*e0) * NN, Wt + (long long)(layer*4 + e0) * DD * DD, h0);
        rgcn_gemm_wmma<<<gemm_grid, 256, 0, stream>>>(
            xs1, rs + (long long)(2*e1) * NN, Wt + (long long)(layer*4 + e1) * DD * DD, h1);
        rgcn_gather_ln<<<node_grid, 256, 0, stream>>>(
            h0, rowptr + (long long)e0 * (NN + 1), csrc + (long long)e0 * EE,
            rs + (long long)(2*e0+1) * NN, bl + (long long)e0 * DD,
            h1, rowptr + (long long)e1 * (NN + 1), csrc + (long long)e1 * EE,
            rs + (long long)(2*e1+1) * NN, bl + (long long)e1 * DD,
            gam, bet, out, NN, act);
    };

    // layer 0 (ReLU per relation), inputs x_A/x_B -> hA1/hB1
    conv_pair(x_A, 0, x_B, 2, 0, b0, lng + (0*2+0)*DD, lnb + (0*2+0)*DD, hA1, 1); // dst A: aa+ba
    conv_pair(x_A, 1, x_B, 3, 0, b0, lng + (0*2+1)*DD, lnb + (0*2+1)*DD, hB1, 1); // dst B: ab+bb

    // layer 1 (no activation), inputs hA1/hB1 -> d_out
    conv_pair(hA1, 0, hB1, 2, 1, b1, lng + (1*2+0)*DD, lnb + (1*2+0)*DD, outA_final, 0);
    conv_pair(hA1, 1, hB1, 3, 1, b1, lng + (1*2+1)*DD, lnb + (1*2+1)*DD, outB_final, 0);
}